// AverageGridEncoder_70257075028518
// MI455X (gfx1250) — compile-verified
//
#include <hip/hip_runtime.h>
#include <cstdint>

// Problem constants (fixed by the reference).
#define B_        4
#define N_        100000
#define P_        128
#define EMBED_    128
#define NUM_GRID_ (P_ * P_)
#define CELLS_PER_BLOCK 8

typedef __attribute__((ext_vector_type(4))) unsigned int v4u;
typedef __attribute__((ext_vector_type(8))) unsigned int v8u;
typedef __attribute__((ext_vector_type(4))) float        v4f;

static constexpr size_t SUMS_ELEMS  = (size_t)B_ * NUM_GRID_ * EMBED_;  // 8,388,608 f32
static constexpr size_t CNT_ELEMS   = (size_t)B_ * NUM_GRID_;           //    65,536 f32
static constexpr size_t XGRID_ELEMS = (size_t)B_ * NUM_GRID_ * 2;       //   131,072 f32
static constexpr size_t NPOINTS     = (size_t)B_ * N_;                  //   400,000

// ---------------------------------------------------------------------------
// Kernel 1: zero the workspace (sums + counts) with b128 stores.
// ---------------------------------------------------------------------------
__global__ void agc_zero_ws(v4f* __restrict__ ws4, size_t n4) {
  size_t i = (size_t)blockIdx.x * blockDim.x + threadIdx.x;
  if (i < n4) ws4[i] = (v4f){0.f, 0.f, 0.f, 0.f};
}

// ---------------------------------------------------------------------------
// Kernel 2: x_grid[b,i,j,:] = (i/127, j/127), written as float2 (b64 stores).
// ---------------------------------------------------------------------------
__global__ void agc_xgrid(float2* __restrict__ out2) {
  int t    = blockIdx.x * blockDim.x + threadIdx.x;   // 0 .. B*NUM_GRID-1
  int cell = t & (NUM_GRID_ - 1);
  int i    = cell >> 7;
  int j    = cell & (P_ - 1);
  const float step = 1.0f / 127.0f;
  out2[t] = make_float2((float)i * step, (float)j * step);
}

// ---------------------------------------------------------------------------
// Kernel 3: per-point nearest-cell index (computed ONCE per point, not once
// per (point, embed) element) + count accumulation. 400k threads total, so
// the two f32 divisions per point are negligible VALU work.
// ---------------------------------------------------------------------------
__global__ void agc_index(const float* __restrict__ x,
                          int* __restrict__ idx,
                          float* __restrict__ cnts) {
  int p = blockIdx.x * blockDim.x + threadIdx.x;
  if (p >= (int)NPOINTS) return;

  const float step = 1.0f / 127.0f;                   // matches f32 (hi-lo)/(p-1)
  float x0 = x[(size_t)p * 2 + 0];
  float x1 = x[(size_t)p * 2 + 1];
  int fi = (int)rintf(x0 / step);                     // rint == round-half-even == jnp.round
  int fj = (int)rintf(x1 / step);
  fi = min(max(fi, 0), P_ - 1);
  fj = min(max(fj, 0), P_ - 1);

  int b    = p / N_;
  int cell = b * NUM_GRID_ + fi * P_ + fj;
  idx[p] = cell;
  atomicAdd(&cnts[cell], 1.0f);
}

// ---------------------------------------------------------------------------
// Kernel 4: scatter-add. One thread per (point, embed) element; 32 lanes of a
// wave share a point -> each atomic instruction is 32 consecutive f32 (128 B
// contiguous) -> GLOBAL_ATOMIC_ADD_F32 fully coalesced into the L2 atomic
// units. z is streamed non-temporally (205 MB read-once) so the 34 MB atomic
// working set (sums) stays resident in the 192 MB L2.
// ---------------------------------------------------------------------------
__global__ void agc_scatter(const float* __restrict__ z,
                            const int* __restrict__ idx,
                            float* __restrict__ sums) {
  size_t gid = (size_t)blockIdx.x * blockDim.x + threadIdx.x;  // exact multiple
  int    e   = (int)(gid & (EMBED_ - 1));
  size_t p   = gid >> 7;                                       // point index

  int   cell = idx[p];                                         // wave-uniform, L0 hit
  float zv   = __builtin_nontemporal_load(z + gid);            // TH=NT stream
  atomicAdd(&sums[(size_t)cell * EMBED_ + e], zv);
}

// ---------------------------------------------------------------------------
// Kernel 5: finalize. Stage this block's latents tile (8 cells * 128 f32 =
// 4 KB) into LDS with the Tensor Data Mover (tensor_load_to_lds), wait on
// TENSORcnt in the issuing wave, barrier, then z_grid = latents + sum/cnt.
// Batch loop is innermost so latents is read exactly once from memory; sums
// reads and z_grid writes are non-temporal (read/write-once streams).
// ---------------------------------------------------------------------------
__global__ void agc_finalize(const float* __restrict__ latents,
                             const float* __restrict__ sums,
                             const float* __restrict__ cnts,
                             float* __restrict__ out) {
  __shared__ __align__(16) float slat[CELLS_PER_BLOCK * EMBED_];  // 4 KB, sole LDS var

  const int base_cell = blockIdx.x * CELLS_PER_BLOCK;

  if (threadIdx.x < 32) {  // wave 0 only: issue one TDM op for the whole block
    uint64_t ga      = (uint64_t)(const void*)(latents + (size_t)base_cell * EMBED_);
    uint32_t lds_off = (uint32_t)(uintptr_t)(void*)&slat[0];  // low 32 bits = LDS byte offset
    const unsigned TD0 = CELLS_PER_BLOCK * EMBED_;            // 1024 elements, data_size=4B

    v4u g0;
    g0[0] = 1u;                                   // count=1, user descriptor
    g0[1] = lds_off;                              // lds_addr (bytes)
    g0[2] = (uint32_t)ga;                         // global_addr[31:0] (tile start)
    g0[3] = (uint32_t)(ga >> 32) | (2u << 30);    // global_addr[56:32] | type=2 ("image")

    v8u g1;
    g1[0] = 0x00020000u;                          // workgroup_mask=0, data_size=2 (4 bytes)
    g1[1] = (TD0 & 0xFFFFu) << 16;                // tensor_dim0[15:0]  @ bits 63:48
    g1[2] = ((TD0 >> 16) & 0xFFFFu) | (1u << 16); // tensor_dim0[31:16] | tensor_dim1=1
    g1[3] = (TD0 & 0xFFFFu) << 16;                // tile_dim0 = 1024   @ bits 127:112
    g1[4] = 0u;                                   // tile_dim1=0 (unused), tile_dim2=0
    g1[5] = TD0;                                  // tensor_dim0_stride[31:0]
    g1[6] = 0u;                                   // stride hi, tensor_dim1_stride lo
    g1[7] = 0u;

    // <= 2D tensor: descriptor groups 2/3 are NULL (2-operand form).
    asm volatile("tensor_load_to_lds %0, %1" :: "s"(g0), "s"(g1) : "memory");
    __builtin_amdgcn_s_wait_tensorcnt(0);
  }
  __syncthreads();

  const int t          = threadIdx.x;       // 0..255 -> one float4 of the LDS tile
  const int cell_local = t >> 5;            // 32 quads per cell
  const int qi         = t & 31;
  const int cell       = base_cell + cell_local;

  const v4f* slat4 = (const v4f*)slat;
  v4f lat = slat4[t];                       // ds_load_b128

  const v4f* sums4 = (const v4f*)sums;
  v4f* out4 = ((v4f*)out) + XGRID_ELEMS / 4;   // z_grid starts after x_grid

#pragma unroll
  for (int b = 0; b < B_; ++b) {
    size_t cidx = (size_t)b * NUM_GRID_ + (size_t)cell;
    float  c    = cnts[cidx];
    v4f    s    = __builtin_nontemporal_load(&sums4[cidx * (EMBED_ / 4) + qi]);
    float  inv  = (c > 0.0f) ? (1.0f / c) : 0.0f;    // cnt>0 -> sum/cnt, else 0
    v4f r = lat + s * inv;
    __builtin_nontemporal_store(r, &out4[cidx * (EMBED_ / 4) + qi]);
  }
}

// ---------------------------------------------------------------------------
extern "C" void kernel_launch(void* const* d_in, const int* in_sizes, int n_in,
                              void* d_out, int out_size, void* d_ws, size_t ws_size,
                              hipStream_t stream) {
  const float* x       = (const float*)d_in[0];   // [B, N, 2]
  const float* z       = (const float*)d_in[1];   // [B, N, EMBED]
  const float* latents = (const float*)d_in[2];   // [P, P, EMBED]
  float* out = (float*)d_out;                     // x_grid ++ z_grid

  float* sums = (float*)d_ws;                     // [B, NUM_GRID, EMBED]
  float* cnts = sums + SUMS_ELEMS;                // [B, NUM_GRID]
  int*   idx  = (int*)(cnts + CNT_ELEMS);         // [B*N] per-point cell index

  // 1) zero workspace (must happen every call; harness does not re-poison)
  size_t zero4 = (SUMS_ELEMS + CNT_ELEMS) / 4;    // exact multiple of 4
  hipLaunchKernelGGL(agc_zero_ws, dim3((unsigned)((zero4 + 255) / 256)), dim3(256),
                     0, stream, (v4f*)d_ws, zero4);

  // 2) x_grid
  hipLaunchKernelGGL(agc_xgrid, dim3((B_ * NUM_GRID_) / 256), dim3(256),
                     0, stream, (float2*)out);

  // 3) per-point cell index + counts
  hipLaunchKernelGGL(agc_index, dim3((unsigned)((NPOINTS + 255) / 256)), dim3(256),
                     0, stream, x, idx, cnts);

  // 4) atomic scatter of z into sums
  size_t total = (size_t)B_ * N_ * EMBED_;        // 51,200,000 = 200,000 * 256
  hipLaunchKernelGGL(agc_scatter, dim3((unsigned)(total / 256)), dim3(256),
                     0, stream, z, idx, sums);

  // 5) finalize: latents (via TDM->LDS) + mean -> z_grid
  hipLaunchKernelGGL(agc_finalize, dim3(NUM_GRID_ / CELLS_PER_BLOCK), dim3(256),
                     0, stream, latents, sums, cnts, out);
}